// GraphClassifier_35192962023827
// MI455X (gfx1250) — compile-verified
//
#include <hip/hip_runtime.h>
#include <hip/hip_bf16.h>

typedef float v2f __attribute__((ext_vector_type(2)));
typedef float v8f __attribute__((ext_vector_type(8)));

#define HID 128
#define NCLS 10
#define NEG_SLOPE 0.2f

// ---------------------------------------------------------------------------
// fp32 hardware atomic add (global_atomic_add_f32)
// ---------------------------------------------------------------------------
__device__ __forceinline__ void atomAddF(float* p, float v) {
#if defined(__HIP_DEVICE_COMPILE__)
    unsafeAtomicAdd(p, v);
#else
    atomicAdd(p, v);
#endif
}

// Order-preserving float->uint encoding for atomicMax on floats.
__device__ __forceinline__ unsigned encodeF(float f) {
    unsigned b = __float_as_uint(f);
    return ((int)b >= 0) ? (b | 0x80000000u) : ~b;
}
__device__ __forceinline__ float decodeF(unsigned u) {
    return (u & 0x80000000u) ? __uint_as_float(u ^ 0x80000000u)
                             : __uint_as_float(~u);
}

// ---------------------------------------------------------------------------
// 0) zero h accumulator, emax (encoded), denom
// ---------------------------------------------------------------------------
__global__ void init_kernel(float* __restrict__ h, unsigned* __restrict__ emax,
                            float* __restrict__ denom, int N) {
    int total = N * (HID + 2);
    for (int i = blockIdx.x * blockDim.x + threadIdx.x; i < total;
         i += gridDim.x * blockDim.x) {
        if (i < N * HID) {
            h[i] = 0.0f;
        } else if (i < N * HID + N) {
            emax[i - N * HID] = 0u;           // < encode of any finite float
        } else {
            denom[i - N * HID - N] = 0.0f;
        }
    }
}

// ---------------------------------------------------------------------------
// 1) x = X @ W   (M x 128) = (M x 128)(128 x 128), fp32 WMMA 16x16x4
//    one wave -> one 16x16 tile; 8 waves per block.
// ---------------------------------------------------------------------------
__global__ void gemm_xw_kernel(const float* __restrict__ X,
                               const float* __restrict__ W,
                               float* __restrict__ Xh, int M) {
    const int lane = threadIdx.x & 31;
    const int wave = threadIdx.x >> 5;
    const int tile = blockIdx.x * 8 + wave;
    const int mTile = tile >> 3;          // 8 N-tiles (128/16)
    const int nTile = tile & 7;
    if (mTile * 16 >= M) return;          // wave-uniform exit
    const int mBase = mTile * 16;
    const int nBase = nTile * 16;

    const int r = lane & 15;              // A-row / B-col / C-col
    const int kOff = (lane >> 4) * 2;     // lanes 16..31 hold K+2,K+3

    v8f acc = {};
    const float* __restrict__ Arow = X + (size_t)(mBase + r) * HID;

    #pragma unroll
    for (int k0 = 0; k0 < HID; k0 += 4) {
        const int ka = k0 + kOff;
        v2f a;
        a.x = Arow[ka];
        a.y = Arow[ka + 1];
        v2f b;
        b.x = W[(size_t)ka * HID + nBase + r];
        b.y = W[(size_t)(ka + 1) * HID + nBase + r];
        acc = __builtin_amdgcn_wmma_f32_16x16x4_f32(false, a, false, b,
                                                    (short)0, acc, false, false);
    }

    const int rowHi = (lane >> 4) * 8;    // C: vgpr i = row i / i+8
    #pragma unroll
    for (int i = 0; i < 8; ++i) {
        Xh[(size_t)(mBase + rowHi + i) * HID + nBase + r] = acc[i];
    }
}

// ---------------------------------------------------------------------------
// 2) alpha_s = x @ a_src, alpha_d = x @ a_dst ; one wave per node
// ---------------------------------------------------------------------------
__global__ void alpha_kernel(const float* __restrict__ x,
                             const float* __restrict__ a_src,
                             const float* __restrict__ a_dst,
                             float* __restrict__ as, float* __restrict__ ad,
                             int N) {
    const int node = (blockIdx.x * blockDim.x + threadIdx.x) >> 5;
    const int lane = threadIdx.x & 31;
    if (node >= N) return;
    const float4 xv = *(const float4*)(x + (size_t)node * HID + lane * 4);
    const float4 sv = *(const float4*)(a_src + lane * 4);
    const float4 dv = *(const float4*)(a_dst + lane * 4);
    float ps = xv.x * sv.x + xv.y * sv.y + xv.z * sv.z + xv.w * sv.w;
    float pd = xv.x * dv.x + xv.y * dv.y + xv.z * dv.z + xv.w * dv.w;
    #pragma unroll
    for (int off = 16; off > 0; off >>= 1) {
        ps += __shfl_xor(ps, off, 32);
        pd += __shfl_xor(pd, off, 32);
    }
    if (lane == 0) { as[node] = ps; ad[node] = pd; }
}

// ---------------------------------------------------------------------------
// 3) pass 1: e = leakyrelu(as[src]+ad[dst]); segment max via encoded atomicMax
//    items: [0,E) real edges, [E,E+N) self loops
// ---------------------------------------------------------------------------
__global__ void edge_max_kernel(const float* __restrict__ as,
                                const float* __restrict__ ad,
                                const long long* __restrict__ src,
                                const long long* __restrict__ dst,
                                float* __restrict__ e_edge,
                                unsigned* __restrict__ emax, int E, int N) {
    const int i = blockIdx.x * blockDim.x + threadIdx.x;
    const int T = E + N;
    if (i >= T) return;
    int s, d;
    if (i < E) { s = (int)src[i]; d = (int)dst[i]; }
    else       { s = d = i - E; }
    float e = as[s] + ad[d];
    e = (e > 0.0f) ? e : NEG_SLOPE * e;
    e_edge[i] = e;
    atomicMax(emax + d, encodeF(e));
}

// ---------------------------------------------------------------------------
// 4) pass 2: ex = exp(e - emax[dst]); denom[dst] += ex
// ---------------------------------------------------------------------------
__global__ void edge_exp_kernel(const long long* __restrict__ dst,
                                float* __restrict__ e_edge,
                                const unsigned* __restrict__ emax,
                                float* __restrict__ denom, int E, int N) {
    const int i = blockIdx.x * blockDim.x + threadIdx.x;
    const int T = E + N;
    if (i >= T) return;
    const int d = (i < E) ? (int)dst[i] : (i - E);
    const float m = decodeF(emax[d]);
    const float ex = __expf(e_edge[i] - m);
    e_edge[i] = ex;
    atomAddF(denom + d, ex);
}

// ---------------------------------------------------------------------------
// 5) pass 3: h[dst] += (ex/denom[dst]) * x[src] ; one wave per edge (float4/lane)
// ---------------------------------------------------------------------------
__global__ void scatter_kernel(const long long* __restrict__ src,
                               const long long* __restrict__ dst,
                               const float* __restrict__ e_edge,
                               const float* __restrict__ denom,
                               const float* __restrict__ x,
                               float* __restrict__ h, int E, int N) {
    const int item = (blockIdx.x * blockDim.x + threadIdx.x) >> 5;
    const int lane = threadIdx.x & 31;
    const int T = E + N;
    if (item >= T) return;
    int s, d;
    if (item < E) { s = (int)src[item]; d = (int)dst[item]; }
    else          { s = d = item - E; }
    const float alpha = e_edge[item] / (denom[d] + 1e-16f);
    const float4 xv = *(const float4*)(x + (size_t)s * HID + lane * 4);
    float* hp = h + (size_t)d * HID + lane * 4;
    atomAddF(hp + 0, alpha * xv.x);
    atomAddF(hp + 1, alpha * xv.y);
    atomAddF(hp + 2, alpha * xv.z);
    atomAddF(hp + 3, alpha * xv.w);
}

// ---------------------------------------------------------------------------
// 6) out = relu(h + bias) @ Wc + bc   via fp32 WMMA (classes padded 10->16)
// ---------------------------------------------------------------------------
__global__ void classifier_kernel(const float* __restrict__ H,
                                  const float* __restrict__ bias,
                                  const float* __restrict__ Wc,
                                  const float* __restrict__ bc,
                                  float* __restrict__ out, int M) {
    const int lane = threadIdx.x & 31;
    const int wave = threadIdx.x >> 5;
    const int mTile = blockIdx.x * 8 + wave;
    if (mTile * 16 >= M) return;          // wave-uniform exit
    const int mBase = mTile * 16;

    const int r = lane & 15;
    const int kOff = (lane >> 4) * 2;

    v8f acc = {};
    const float* __restrict__ Hrow = H + (size_t)(mBase + r) * HID;
    const bool valid = (r < NCLS);

    #pragma unroll
    for (int k0 = 0; k0 < HID; k0 += 4) {
        const int ka = k0 + kOff;
        float h0 = Hrow[ka] + bias[ka];
        float h1 = Hrow[ka + 1] + bias[ka + 1];
        v2f a;
        a.x = (h0 > 0.0f) ? h0 : 0.0f;    // fused ReLU
        a.y = (h1 > 0.0f) ? h1 : 0.0f;
        v2f b;
        b.x = valid ? Wc[ka * NCLS + r] : 0.0f;
        b.y = valid ? Wc[(ka + 1) * NCLS + r] : 0.0f;
        acc = __builtin_amdgcn_wmma_f32_16x16x4_f32(false, a, false, b,
                                                    (short)0, acc, false, false);
    }

    if (valid) {
        const float bcv = bc[r];
        const int rowHi = (lane >> 4) * 8;
        #pragma unroll
        for (int i = 0; i < 8; ++i) {
            const int row = mBase + rowHi + i;
            if (row < M) out[(size_t)row * NCLS + r] = acc[i] + bcv;
        }
    }
}

// ---------------------------------------------------------------------------
// launch
// ---------------------------------------------------------------------------
extern "C" void kernel_launch(void* const* d_in, const int* in_sizes, int n_in,
                              void* d_out, int out_size, void* d_ws, size_t ws_size,
                              hipStream_t stream) {
    const float*     X     = (const float*)d_in[0];
    const long long* edges = (const long long*)d_in[1];
    const float*     W     = (const float*)d_in[2];
    const float*     a_src = (const float*)d_in[3];
    const float*     a_dst = (const float*)d_in[4];
    const float*     bias  = (const float*)d_in[5];
    const float*     Wc    = (const float*)d_in[6];
    const float*     bc    = (const float*)d_in[7];
    float* out = (float*)d_out;

    const int N = in_sizes[0] / HID;
    const int E = in_sizes[1] / 2;
    const int T = E + N;

    const long long* src = edges;
    const long long* dst = edges + E;

    // workspace carve-up
    char* ws = (char*)d_ws;
    float*    x      = (float*)ws;                       ws += (size_t)N * HID * 4;
    float*    h      = (float*)ws;                       ws += (size_t)N * HID * 4;
    float*    as     = (float*)ws;                       ws += (size_t)N * 4;
    float*    ad     = (float*)ws;                       ws += (size_t)N * 4;
    unsigned* emax   = (unsigned*)ws;                    ws += (size_t)N * 4;
    float*    denom  = (float*)ws;                       ws += (size_t)N * 4;
    float*    e_edge = (float*)ws;

    const int BLK = 256;

    // 0) zero accumulators
    {
        int total = N * (HID + 2);
        int grid = (total + BLK - 1) / BLK;
        if (grid > 65535 * 8) grid = 65535 * 8;
        init_kernel<<<grid, BLK, 0, stream>>>(h, emax, denom, N);
    }
    // 1) x = X @ W
    {
        int mTiles = (N + 15) / 16;
        int tiles = mTiles * 8;
        gemm_xw_kernel<<<(tiles + 7) / 8, BLK, 0, stream>>>(X, W, x, N);
    }
    // 2) alpha projections
    alpha_kernel<<<(N + 7) / 8, BLK, 0, stream>>>(x, a_src, a_dst, as, ad, N);
    // 3) segment max
    edge_max_kernel<<<(T + BLK - 1) / BLK, BLK, 0, stream>>>(as, ad, src, dst,
                                                             e_edge, emax, E, N);
    // 4) exp + denom
    edge_exp_kernel<<<(T + BLK - 1) / BLK, BLK, 0, stream>>>(dst, e_edge, emax,
                                                             denom, E, N);
    // 5) weighted scatter
    scatter_kernel<<<(T + 7) / 8, BLK, 0, stream>>>(src, dst, e_edge, denom,
                                                    x, h, E, N);
    // 6) classifier
    {
        int mTiles = (N + 15) / 16;
        classifier_kernel<<<(mTiles + 7) / 8, BLK, 0, stream>>>(h, bias, Wc, bc,
                                                                out, N);
    }
}